// ScatterND_8890582303351
// MI455X (gfx1250) — compile-verified
//
#include <hip/hip_runtime.h>
#include <hip/hip_bf16.h>

typedef float float4v __attribute__((ext_vector_type(4)));

// ---------------------------------------------------------------------------
// Pass 1a: flag[row] = -1 for all rows
// ---------------------------------------------------------------------------
__global__ void scat_init_flags(int* __restrict__ flags, long n) {
    long i = (long)blockIdx.x * blockDim.x + threadIdx.x;
    if (i < n) flags[i] = -1;
}

// ---------------------------------------------------------------------------
// Pass 1b: flag[idx[k]] = k   (indices are unique per the reference)
// ---------------------------------------------------------------------------
__global__ void scat_scatter_flags(const long long* __restrict__ idx,
                                   int* __restrict__ flags, long k) {
    long i = (long)blockIdx.x * blockDim.x + threadIdx.x;
    if (i < k) flags[(long)idx[i]] = (int)i;
}

// ---------------------------------------------------------------------------
// Pass 2 (fused, D == 128): one wave32 per row; lane l owns floats [4l, 4l+4).
// Streaming data: non-temporal b128 loads/stores (512MB/128MB >> 192MB L2).
// flag[] (4MB) uses regular-temporal loads so it can live in L2.
// Flag is wave-uniform: load in lane 0, broadcast via readfirstlane so the
// "row is scattered?" test is a scalar branch (75% of waves skip the load).
// ---------------------------------------------------------------------------
__global__ void scat_fused_d128(const float* __restrict__ data,
                                const float* __restrict__ upd,
                                const int* __restrict__ flags,
                                float* __restrict__ out, long nrows) {
    long wave = ((long)blockIdx.x * blockDim.x + threadIdx.x) >> 5;
    int  lane = threadIdx.x & 31;
    if (wave >= nrows) return;

    long base = wave * 128 + (long)lane * 4;
    float4v v = __builtin_nontemporal_load((const float4v*)(data + base));

    int f = 0;
    if (lane == 0) f = flags[wave];
    f = __builtin_amdgcn_readfirstlane(f);   // wave-uniform -> SGPR -> scalar branch

    if (f >= 0) {
        float4v u = __builtin_nontemporal_load(
            (const float4v*)(upd + (long)f * 128 + (long)lane * 4));
        v += u;
    }
    __builtin_nontemporal_store(v, (float4v*)(out + base));
}

// ---------------------------------------------------------------------------
// Fallback path (generic D, or workspace too small): copy then RMW.
// Valid because indices are unique (no collisions -> no atomics needed).
// ---------------------------------------------------------------------------
__global__ void scat_copy(const float* __restrict__ src,
                          float* __restrict__ dst, long total4) {
    long i = (long)blockIdx.x * blockDim.x + threadIdx.x;
    long stride = (long)gridDim.x * blockDim.x;
    for (; i < total4; i += stride) {
        float4v v = __builtin_nontemporal_load((const float4v*)src + i);
        __builtin_nontemporal_store(v, (float4v*)dst + i);
    }
}

__global__ void scat_rmw(const long long* __restrict__ idx,
                         const float* __restrict__ upd,
                         float* __restrict__ out, long k, long d) {
    long i = (long)blockIdx.x * blockDim.x + threadIdx.x;
    long total = k * d;
    if (i >= total) return;
    long kk = i / d;
    long dd = i - kk * d;
    long row = (long)idx[kk];
    out[row * d + dd] += upd[i];
}

// ---------------------------------------------------------------------------
extern "C" void kernel_launch(void* const* d_in, const int* in_sizes, int n_in,
                              void* d_out, int out_size, void* d_ws, size_t ws_size,
                              hipStream_t stream) {
    (void)n_in; (void)out_size;
    const float*     data = (const float*)d_in[0];
    const long long* idx  = (const long long*)d_in[1];
    const float*     upd  = (const float*)d_in[2];
    float*           out  = (float*)d_out;

    long K = (long)in_sizes[1];                     // indices: (K, 1)
    long D = (K > 0) ? (long)in_sizes[2] / K : 1;   // updates: (K, D)
    long N = (D > 0) ? (long)in_sizes[0] / D : 0;   // data:    (N, D)

    const int BLK = 256;

    bool fast = (D == 128) && (ws_size >= (size_t)N * sizeof(int));
    if (fast) {
        int* flags = (int*)d_ws;

        long gInit = (N + BLK - 1) / BLK;
        scat_init_flags<<<(unsigned)gInit, BLK, 0, stream>>>(flags, N);

        long gScat = (K + BLK - 1) / BLK;
        scat_scatter_flags<<<(unsigned)gScat, BLK, 0, stream>>>(idx, flags, K);

        // 8 waves (8 rows) per 256-thread block
        long rowsPerBlk = BLK / 32;
        long gFused = (N + rowsPerBlk - 1) / rowsPerBlk;
        scat_fused_d128<<<(unsigned)gFused, BLK, 0, stream>>>(data, upd, flags, out, N);
    } else {
        long total = N * D;
        if ((total & 3) == 0) {
            long total4 = total >> 2;
            long g = (total4 + BLK - 1) / BLK;
            if (g > 262144) g = 262144;   // grid-stride
            scat_copy<<<(unsigned)g, BLK, 0, stream>>>(data, out, total4);
        } else {
            hipMemcpyAsync(out, data, total * sizeof(float), hipMemcpyDeviceToDevice, stream);
        }
        long totalKD = K * D;
        long g2 = (totalKD + BLK - 1) / BLK;
        scat_rmw<<<(unsigned)g2, BLK, 0, stream>>>(idx, upd, out, K, D);
    }
}